// PositionalAttention_87840671138013
// MI455X (gfx1250) — compile-verified
//
#include <hip/hip_runtime.h>
#include <hip/hip_bf16.h>

typedef __attribute__((ext_vector_type(2))) float v2f;
typedef __attribute__((ext_vector_type(8))) float v8f;

#define BATCH 4
#define DEC 256
#define ENC 256
#define HDIM 512

// ---------------------------------------------------------------------------
// fp32 WMMA helpers: D(16x16) += A(16x4) * B(4x16), per ISA 7.12.2 layouts.
// A: lane%16 = M row, (lane/16)*2 + vgpr = K.   C/D: lane%16 = N, vgpr = M
// (lanes 16-31 hold M+8). B assumed mirror of A with N in place of M.
// ---------------------------------------------------------------------------

// Kernel 1: EP[b,e,m] = enc[b,e,:] @ W_mlp[0:512, m]
//           DP[b,d,m] = dec[b,d,:] @ W_mlp[512:1024, m]
// One wave per 16x16 output tile. Tiles: src(2) x b(4) x mt(16) x nt(32) = 4096.
__global__ __launch_bounds__(256)
void proj_wmma_kernel(const float* __restrict__ dec_s,
                      const float* __restrict__ enc_s,
                      const float* __restrict__ Wmlp,
                      float* __restrict__ EP,
                      float* __restrict__ DP) {
    const int lane = threadIdx.x & 31;
    const int wid  = threadIdx.x >> 5;
    int w = blockIdx.x * 8 + wid;

    const int src = w >> 11;           // 0 = enc-proj, 1 = dec-proj
    int rem = w & 2047;
    const int b  = rem >> 9;
    rem &= 511;
    const int m0 = (rem >> 5) << 4;    // row tile (within 256)
    const int n0 = (rem & 31) << 4;    // col tile (within 512)

    const float* A = (src == 0 ? enc_s : dec_s) + (size_t)b * DEC * HDIM;
    const float* W = Wmlp + (size_t)(src == 0 ? 0 : HDIM) * HDIM;
    float*       O = (src == 0 ? EP : DP) + (size_t)b * DEC * HDIM;

    const int row   = m0 + (lane & 15);
    const int col   = n0 + (lane & 15);
    const int khalf = (lane >> 4) << 1;        // 0 or 2

    const float* Arow = A + (size_t)row * HDIM;
    v8f acc = {};
#pragma unroll 4
    for (int k0 = 0; k0 < HDIM; k0 += 4) {
        v2f a = *(const v2f*)(Arow + k0 + khalf);            // 8B aligned (khalf even)
        const float* Wp = W + (size_t)(k0 + khalf) * HDIM + col;
        v2f bb;
        bb[0] = Wp[0];
        bb[1] = Wp[HDIM];
        acc = __builtin_amdgcn_wmma_f32_16x16x4_f32(false, a, false, bb,
                                                    (short)0, acc, false, false);
    }

    const int rbase = m0 + ((lane >> 4) << 3);
    float* Op = O + (size_t)rbase * HDIM + col;
#pragma unroll
    for (int v = 0; v < 8; ++v) Op[(size_t)v * HDIM] = acc[v];
}

// ---------------------------------------------------------------------------
// Kernel 2: scores + masks + softmax.  Block = 256 threads = one (b, 8-d tile);
// thread t owns column e = t.  enc_proj chunk staged in LDS transposed with
// pitch 257 (odd mod 64 -> conflict-free strided writes AND unit-stride reads).
// ---------------------------------------------------------------------------
#define DT 8       // d rows per block
#define MC 32      // m-chunk
#define EPAD 257   // LDS pitch for the e dimension

__device__ inline float fast_tanh(float x) {
#if __has_builtin(__builtin_amdgcn_tanhf)
    return __builtin_amdgcn_tanhf(x);      // v_tanh_f32 on gfx1250
#else
    float e = __expf(2.0f * x);            // v_exp_f32
    return 1.0f - 2.0f / (e + 1.0f);       // exact limits at +/-inf
#endif
}

__global__ __launch_bounds__(256)
void score_softmax_kernel(const float* __restrict__ EP,
                          const float* __restrict__ DP,
                          const float* __restrict__ enc_s,
                          const unsigned char* __restrict__ extmask,
                          const float* __restrict__ bmlp,
                          const float* __restrict__ Wout,
                          const float* __restrict__ bOut,
                          float* __restrict__ attn) {
    __shared__ float EPs[MC * EPAD];   // [mi][e], pitch 257
    __shared__ float DPs[DT * MC];     // bias folded in
    __shared__ float wouts[MC];
    __shared__ float red[8 * DT];      // per-wave partials for softmax

    const int tid = threadIdx.x;
    const int b   = blockIdx.x >> 5;             // 32 d-tiles per batch
    const int d0  = (blockIdx.x & 31) * DT;
    const int e   = tid;                         // ENC == blockDim.x

    const float* EPb = EP + (size_t)b * ENC * HDIM;
    const float* DPb = DP + (size_t)b * DEC * HDIM;

    float s[DT];
#pragma unroll
    for (int d = 0; d < DT; ++d) s[d] = 0.0f;

    for (int mc = 0; mc < HDIM; mc += MC) {
        __syncthreads();
        // DP tile + bias: 8x32 = 256 elems, one per thread
        {
            int d = tid >> 5, mi = tid & 31;
            DPs[tid] = DPb[(size_t)(d0 + d) * HDIM + mc + mi] + bmlp[mc + mi];
        }
        if (tid < MC) wouts[tid] = Wout[mc + tid];
        // EP chunk transposed: 256e x 32m, coalesced global reads (mi fastest)
#pragma unroll 4
        for (int it = 0; it < (ENC * MC) / 256; ++it) {
            int f  = it * 256 + tid;
            int ee = f >> 5, mi = f & 31;
            EPs[mi * EPAD + ee] = EPb[(size_t)ee * HDIM + mc + mi];
        }
        __syncthreads();

        for (int mi = 0; mi < MC; ++mi) {
            float ep = EPs[mi * EPAD + e];       // unit stride across lanes
            float wv = wouts[mi];
#pragma unroll
            for (int d = 0; d < DT; ++d) {
                float x = DPs[d * MC + mi] + ep; // broadcast LDS read
                s[d] = __builtin_fmaf(fast_tanh(x), wv, s[d]);
            }
        }
    }

    // masks
    const float bo   = bOut[0];
    const bool  padm = (enc_s[((size_t)b * ENC + e) * HDIM] == 0.0f);
    const int lane = tid & 31, wv = tid >> 5;

    float st[DT];
#pragma unroll
    for (int d = 0; d < DT; ++d) {
        float v = s[d] + bo;
        if (extmask[(size_t)(b * DEC + d0 + d) * ENC + e]) v = -__builtin_inff();
        if (padm) v = -__builtin_inff();
        st[d] = v;
        float m = v;
        for (int off = 16; off; off >>= 1) m = fmaxf(m, __shfl_xor(m, off));
        if (lane == 0) red[wv * DT + d] = m;
    }
    __syncthreads();
    float rowmax[DT];
#pragma unroll
    for (int d = 0; d < DT; ++d) {
        float m = red[d];
        for (int ww = 1; ww < 8; ++ww) m = fmaxf(m, red[ww * DT + d]);
        rowmax[d] = m;
    }
    __syncthreads();
    float ex[DT];
#pragma unroll
    for (int d = 0; d < DT; ++d) {
        ex[d] = __expf(st[d] - rowmax[d]);
        float t = ex[d];
        for (int off = 16; off; off >>= 1) t += __shfl_xor(t, off);
        if (lane == 0) red[wv * DT + d] = t;
    }
    __syncthreads();
#pragma unroll
    for (int d = 0; d < DT; ++d) {
        float tot = red[d];
        for (int ww = 1; ww < 8; ++ww) tot += red[ww * DT + d];
        attn[(size_t)(b * DEC + d0 + d) * ENC + e] = ex[d] / tot;
    }
}

// ---------------------------------------------------------------------------
// Kernel 3: context[b] = attn[b](256x256) @ enc[b](256x512), fp32 WMMA.
// Tiles: b(4) x mt(16) x nt(32) = 2048 waves.
// ---------------------------------------------------------------------------
__global__ __launch_bounds__(256)
void context_wmma_kernel(const float* __restrict__ attn,
                         const float* __restrict__ enc_s,
                         float* __restrict__ ctx) {
    const int lane = threadIdx.x & 31;
    const int wid  = threadIdx.x >> 5;
    int w = blockIdx.x * 8 + wid;
    const int b  = w >> 9;
    int rem = w & 511;
    const int m0 = (rem >> 5) << 4;
    const int n0 = (rem & 31) << 4;

    const float* A  = attn  + (size_t)b * DEC * ENC;
    const float* Bm = enc_s + (size_t)b * ENC * HDIM;
    float*       O  = ctx   + (size_t)b * DEC * HDIM;

    const int row   = m0 + (lane & 15);
    const int col   = n0 + (lane & 15);
    const int khalf = (lane >> 4) << 1;

    const float* Arow = A + (size_t)row * ENC;
    v8f acc = {};
#pragma unroll 4
    for (int k0 = 0; k0 < ENC; k0 += 4) {
        v2f a = *(const v2f*)(Arow + k0 + khalf);
        const float* Bp = Bm + (size_t)(k0 + khalf) * HDIM + col;
        v2f bb;
        bb[0] = Bp[0];
        bb[1] = Bp[HDIM];
        acc = __builtin_amdgcn_wmma_f32_16x16x4_f32(false, a, false, bb,
                                                    (short)0, acc, false, false);
    }
    const int rbase = m0 + ((lane >> 4) << 3);
    float* Op = O + (size_t)rbase * HDIM + col;
#pragma unroll
    for (int v = 0; v < 8; ++v) Op[(size_t)v * HDIM] = acc[v];
}

// ---------------------------------------------------------------------------
extern "C" void kernel_launch(void* const* d_in, const int* in_sizes, int n_in,
                              void* d_out, int out_size, void* d_ws, size_t ws_size,
                              hipStream_t stream) {
    const float*         dec_s   = (const float*)d_in[0];
    const float*         enc_s   = (const float*)d_in[1];
    const unsigned char* extmask = (const unsigned char*)d_in[2];   // bool mask
    const float*         Wmlp    = (const float*)d_in[3];
    const float*         bmlp    = (const float*)d_in[4];
    const float*         Wout    = (const float*)d_in[5];
    const float*         bOut    = (const float*)d_in[6];

    float* out  = (float*)d_out;
    float* ctx  = out;                                  // [B, DEC, H]
    float* attn = out + (size_t)BATCH * DEC * HDIM;     // [B, DEC, ENC]

    float* EP = (float*)d_ws;                           // [B, ENC, H]
    float* DP = EP + (size_t)BATCH * ENC * HDIM;        // [B, DEC, H]

    // 1) projections (fp32 WMMA): 4096 wave-tiles / 8 waves per block
    proj_wmma_kernel<<<512, 256, 0, stream>>>(dec_s, enc_s, Wmlp, EP, DP);
    // 2) tanh-score + masked softmax: one block per (b, 8 decoder rows)
    score_softmax_kernel<<<BATCH * (DEC / DT), 256, 0, stream>>>(
        EP, DP, enc_s, extmask, bmlp, Wout, bOut, attn);
    // 3) context GEMM (fp32 WMMA): 2048 wave-tiles / 8 per block
    context_wmma_kernel<<<256, 256, 0, stream>>>(attn, enc_s, ctx);
}